// NonLocalMP_38972533243937
// MI455X (gfx1250) — compile-verified
//
#include <hip/hip_runtime.h>

// ---------------------------------------------------------------------------
// NonLocalMP on MI455X (gfx1250): fp32 WMMA (v_wmma_f32_16x16x4_f32) pipeline.
// B=16, N=64, C=512.  Dominant GEMM: per (b,i) block, [64,512]x[512,512].
// ---------------------------------------------------------------------------

typedef __attribute__((ext_vector_type(2))) float v2f;
typedef __attribute__((ext_vector_type(8))) float v8f;

#define C_DIM 512
#define N_PTS 64
#define B_SZ 16
#define R_ROWS (B_SZ * N_PTS) /* 1024 */
#define LDH 516               /* padded LDS row stride (floats): 516%64=4 -> no bank conflicts */
#define NEG_SLOPE 0.2f
#define BN_EPS 1e-5f

__device__ __forceinline__ float leaky(float x) { return x >= 0.0f ? x : NEG_SLOPE * x; }

// D = A(16x4 f32) * B(4x16 f32) + C(16x16 f32), wave32.
// A frag: lane<16 -> (M=lane, K=k+0..1); lane>=16 -> (M=lane-16, K=k+2..3)
// B frag: lane<16 -> (K=k+0..1, N=lane); lane>=16 -> (K=k+2..3, N=lane-16)
// C/D:    reg r: lane<16 -> (M=r, N=lane); lane>=16 -> (M=r+8, N=lane-16)
__device__ __forceinline__ v8f wmma_k4(v2f a, v2f b, v8f c) {
  return __builtin_amdgcn_wmma_f32_16x16x4_f32(false, a, false, b, (short)0, c,
                                               false, false);
}

// ---- repack W[K][Nn] row-major -> Wp[k2][n] = (W[2k2][n], W[2k2+1][n]) ----
__global__ void __launch_bounds__(256)
pack_w(const float* __restrict__ W, v2f* __restrict__ Wp, int K, int Nn) {
  int idx = blockIdx.x * 256 + threadIdx.x;
  int total = (K >> 1) * Nn;
  if (idx >= total) return;
  int k2 = idx / Nn;
  int n = idx - k2 * Nn;
  v2f v;
  v.x = W[(size_t)(2 * k2) * Nn + n];
  v.y = W[(size_t)(2 * k2 + 1) * Nn + n];
  Wp[idx] = v;
}

// ---- Out[1024,512] = X[1024,512] @ Wp (K=512), one 16x16 tile per wave ----
__global__ void __launch_bounds__(256)
gemm_wmma(const float* __restrict__ X, const v2f* __restrict__ Wp,
          float* __restrict__ Out) {
  const int lane = threadIdx.x & 31;
  const int wave = threadIdx.x >> 5;
  const int t = blockIdx.x * 8 + wave;         // 2048 tiles
  const int m0 = (t >> 5) << 4;                // 32 n-tiles per row band
  const int n0 = (t & 31) << 4;
  const int half = lane >> 4;
  const int l16 = lane & 15;
  const int arow = m0 + l16;
  const int n = n0 + l16;

  v8f acc = {};
  for (int k = 0; k < C_DIM; k += 4) {
    const int kk = k + (half << 1);
    v2f a = *(const v2f*)(X + (size_t)arow * C_DIM + kk);
    v2f b = Wp[(size_t)(kk >> 1) * C_DIM + n];
    acc = wmma_k4(a, b, acc);
  }
#pragma unroll
  for (int r = 0; r < 8; ++r)
    Out[(size_t)(m0 + r + 8 * half) * C_DIM + n] = acc[r];
}

// ---- per (b,i): h=leaky(A_i + Bp_j + b1) in LDS; M_i = sum_j leaky(h@W2+b2) ----
__global__ void __launch_bounds__(256)
relation_kernel(const float* __restrict__ Abuf, const float* __restrict__ Bbuf,
                const float* __restrict__ b1, const float* __restrict__ b2,
                const v2f* __restrict__ Wp2, float* __restrict__ Mout) {
  extern __shared__ float hsm[];               // [64][LDH], ~129 KB
  const int bi = blockIdx.x;                   // b*64 + i
  const int bbase = bi & ~(N_PTS - 1);         // b*64
  const float* Arow = Abuf + (size_t)bi * C_DIM;

  // Build h tile [64 x 512] cooperatively (float4 vectorized).
  for (int idx = threadIdx.x; idx < N_PTS * (C_DIM / 4); idx += 256) {
    int j = idx >> 7;                          // idx / 128
    int c4 = (idx & 127) << 2;
    float4 av = *(const float4*)(Arow + c4);
    float4 bv = *(const float4*)(Bbuf + (size_t)(bbase + j) * C_DIM + c4);
    float4 b1v = *(const float4*)(b1 + c4);
    float4 hv;
    hv.x = leaky(av.x + bv.x + b1v.x);
    hv.y = leaky(av.y + bv.y + b1v.y);
    hv.z = leaky(av.z + bv.z + b1v.z);
    hv.w = leaky(av.w + bv.w + b1v.w);
    *(float4*)(hsm + (size_t)j * LDH + c4) = hv;
  }
  __syncthreads();

  const int lane = threadIdx.x & 31;
  const int wave = threadIdx.x >> 5;
  const int half = lane >> 4;
  const int l16 = lane & 15;

  // 8 waves x 4 strips = 32 n-tiles (full 512 cols). Each strip: 4 M-tiles
  // (all 64 j rows) so the W2 fragment is loaded once per k-step.
  for (int s = 0; s < 4; ++s) {
    const int n0 = ((wave << 2) + s) << 4;
    const int n = n0 + l16;
    v8f c0 = {}, c1 = {}, c2 = {}, c3 = {};
    for (int k = 0; k < C_DIM; k += 4) {
      const int kk = k + (half << 1);
      v2f bf2 = Wp2[(size_t)(kk >> 1) * C_DIM + n];
      v2f f0 = *(const v2f*)(hsm + (size_t)(l16) * LDH + kk);
      v2f f1 = *(const v2f*)(hsm + (size_t)(16 + l16) * LDH + kk);
      v2f f2 = *(const v2f*)(hsm + (size_t)(32 + l16) * LDH + kk);
      v2f f3 = *(const v2f*)(hsm + (size_t)(48 + l16) * LDH + kk);
      c0 = wmma_k4(f0, bf2, c0);
      c1 = wmma_k4(f1, bf2, c1);
      c2 = wmma_k4(f2, bf2, c2);
      c3 = wmma_k4(f3, bf2, c3);
    }
    // epilogue: +b2, leaky, reduce over the 64 j rows of this column strip
    const float b2v = b2[n];
    float cs = 0.0f;
#pragma unroll
    for (int r = 0; r < 8; ++r) {
      cs += leaky(c0[r] + b2v);
      cs += leaky(c1[r] + b2v);
      cs += leaky(c2[r] + b2v);
      cs += leaky(c3[r] + b2v);
    }
    cs += __shfl_xor(cs, 16, 32);              // lanes L and L+16 hold same N
    if (half == 0) Mout[(size_t)bi * C_DIM + n] = cs;
  }
}

// ---- Y[1024,512] = [F | M] @ Wpf (K=1024) + bf ----
__global__ void __launch_bounds__(256)
fusion_gemm(const float* __restrict__ F, const float* __restrict__ Mb,
            const v2f* __restrict__ Wpf, const float* __restrict__ bf,
            float* __restrict__ Y) {
  const int lane = threadIdx.x & 31;
  const int wave = threadIdx.x >> 5;
  const int t = blockIdx.x * 8 + wave;
  const int m0 = (t >> 5) << 4;
  const int n0 = (t & 31) << 4;
  const int half = lane >> 4;
  const int l16 = lane & 15;
  const int arow = m0 + l16;
  const int n = n0 + l16;

  v8f acc = {};
  for (int k = 0; k < 2 * C_DIM; k += 4) {
    const int kk = k + (half << 1);
    // concat([F, M]) along K; boundary at 512 never splits a k-step pair
    const float* src = (kk < C_DIM) ? (F + (size_t)arow * C_DIM + kk)
                                    : (Mb + (size_t)arow * C_DIM + (kk - C_DIM));
    v2f a = *(const v2f*)src;
    v2f b = Wpf[(size_t)(kk >> 1) * C_DIM + n];
    acc = wmma_k4(a, b, acc);
  }
  const float bfv = bf[n];
#pragma unroll
  for (int r = 0; r < 8; ++r)
    Y[(size_t)(m0 + r + 8 * half) * C_DIM + n] = acc[r] + bfv;
}

// ---- deterministic column stats (sum, sumsq) in two stages ----
__global__ void __launch_bounds__(256)
colstats_part(const float* __restrict__ Y, float* __restrict__ part) {
  int tid = blockIdx.x * 256 + threadIdx.x;    // 8192 threads
  int seg = tid >> 9;                          // 16 row segments of 64 rows
  int n = tid & (C_DIM - 1);
  const float* p = Y + (size_t)seg * 64 * C_DIM + n;
  float s1 = 0.0f, s2 = 0.0f;
  for (int r = 0; r < 64; ++r) {
    float y = p[(size_t)r * C_DIM];
    s1 += y;
    s2 += y * y;
  }
  part[seg * 2 * C_DIM + n] = s1;
  part[seg * 2 * C_DIM + C_DIM + n] = s2;
}

__global__ void __launch_bounds__(256)
colstats_final(const float* __restrict__ part, float* __restrict__ sums) {
  int n = blockIdx.x * 256 + threadIdx.x;      // 512 threads
  if (n >= C_DIM) return;
  float s1 = 0.0f, s2 = 0.0f;
  for (int seg = 0; seg < 16; ++seg) {
    s1 += part[seg * 2 * C_DIM + n];
    s2 += part[seg * 2 * C_DIM + C_DIM + n];
  }
  sums[n] = s1;
  sums[C_DIM + n] = s2;
}

// ---- BN(train stats) + gamma/beta + leaky ----
__global__ void __launch_bounds__(256)
bn_apply(const float* __restrict__ Y, const float* __restrict__ sums,
         const float* __restrict__ gamma, const float* __restrict__ beta,
         float* __restrict__ out) {
  int idx = blockIdx.x * 256 + threadIdx.x;    // 1024*512
  int n = idx & (C_DIM - 1);
  const float inv = 1.0f / (float)R_ROWS;
  float mean = sums[n] * inv;
  float var = sums[C_DIM + n] * inv - mean * mean;
  float rs = rsqrtf(var + BN_EPS);
  float y = (Y[idx] - mean) * rs * gamma[n] + beta[n];
  out[idx] = leaky(y);
}

// ---------------------------------------------------------------------------
// Workspace layout (float offsets); total ~13.8 MB.
// ---------------------------------------------------------------------------
#define WP1_OFF 0u           /* 262144 v2f = 524288 f : packed W1 (K=1024)   */
#define WP2_OFF 524288u      /* 131072 v2f            : packed W2 (K=512)    */
#define WPF_OFF 786432u      /* 262144 v2f            : packed Wf (K=1024)   */
#define A_OFF 1310720u       /* 1024x512 : F @ W1[:C]                        */
#define BP_OFF 1835008u      /* 1024x512 : F @ W1[C:]                        */
#define M_OFF 2359296u       /* 1024x512 : relation output                   */
#define Y_OFF 2883584u       /* 1024x512 : fusion pre-BN                     */
#define SUMS_OFF 3407872u    /* 1024 : col sum / sumsq                       */
#define PART_OFF 3408896u    /* 16*1024 : partial stats                      */

extern "C" void kernel_launch(void* const* d_in, const int* in_sizes, int n_in,
                              void* d_out, int out_size, void* d_ws,
                              size_t ws_size, hipStream_t stream) {
  (void)in_sizes; (void)n_in; (void)out_size; (void)ws_size;

  const float* F = (const float*)d_in[0];
  const float* W1 = (const float*)d_in[1];
  const float* b1 = (const float*)d_in[2];
  const float* W2 = (const float*)d_in[3];
  const float* b2 = (const float*)d_in[4];
  const float* Wf = (const float*)d_in[5];
  const float* bf = (const float*)d_in[6];
  const float* gamma = (const float*)d_in[7];
  const float* beta = (const float*)d_in[8];
  float* out = (float*)d_out;
  float* ws = (float*)d_ws;

  v2f* Wp1 = (v2f*)(ws + WP1_OFF);
  v2f* Wp2 = (v2f*)(ws + WP2_OFF);
  v2f* Wpf = (v2f*)(ws + WPF_OFF);
  float* Abuf = ws + A_OFF;
  float* Bbuf = ws + BP_OFF;
  float* Mbuf = ws + M_OFF;
  float* Ybuf = ws + Y_OFF;
  float* sums = ws + SUMS_OFF;
  float* part = ws + PART_OFF;

  // 1) repack weights for contiguous B-fragment loads
  pack_w<<<1024, 256, 0, stream>>>(W1, Wp1, 2 * C_DIM, C_DIM);
  pack_w<<<512, 256, 0, stream>>>(W2, Wp2, C_DIM, C_DIM);
  pack_w<<<1024, 256, 0, stream>>>(Wf, Wpf, 2 * C_DIM, C_DIM);

  // 2) A = F@W1[:C], Bp = F@W1[C:]   (2048 tiles / 8 waves = 256 blocks)
  gemm_wmma<<<256, 256, 0, stream>>>(F, Wp1, Abuf);
  gemm_wmma<<<256, 256, 0, stream>>>(F, Wp1 + (size_t)256 * C_DIM, Bbuf);

  // 3) relation MLP + j-reduction, one workgroup per (b,i); h lives in LDS
  relation_kernel<<<R_ROWS, 256, N_PTS * LDH * sizeof(float), stream>>>(
      Abuf, Bbuf, b1, b2, Wp2, Mbuf);

  // 4) fusion linear on [F|M]
  fusion_gemm<<<256, 256, 0, stream>>>(F, Mbuf, Wpf, bf, Ybuf);

  // 5) BN stats (deterministic two-stage) + apply + leaky
  colstats_part<<<32, 256, 0, stream>>>(Ybuf, part);
  colstats_final<<<2, 256, 0, stream>>>(part, sums);
  bn_apply<<<2048, 256, 0, stream>>>(Ybuf, sums, gamma, beta, out);
}